// KLayerHeteroRGCN_30133490549165
// MI455X (gfx1250) — compile-verified
//
#include <hip/hip_runtime.h>
#include <math.h>

#define N_NODES 50000
#define N_REL 4
#define N_EDGES 800000
#define F 128
#define N_CLASSES 16

typedef float v2f __attribute__((ext_vector_type(2)));
typedef float v8f __attribute__((ext_vector_type(8)));

static __device__ __forceinline__ void atomic_add_f32(float* p, float v) {
    __hip_atomic_fetch_add(p, v, __ATOMIC_RELAXED, __HIP_MEMORY_SCOPE_AGENT);
}

// ---------------------------------------------------------------- zero
__global__ __launch_bounds__(256) void zero_kernel(float* __restrict__ p, int n) {
    int i = blockIdx.x * 256 + threadIdx.x;
    if (i < n) p[i] = 0.0f;
}

// ---------------------------------------------------------------- degree count (all relations at once)
__global__ __launch_bounds__(256)
void degree_kernel(float* __restrict__ cnt_out, float* __restrict__ cnt_in,
                   const int* __restrict__ src, const int* __restrict__ dst, int total) {
    int i = blockIdx.x * 256 + threadIdx.x;
    if (i >= total) return;
    int r = i / N_EDGES;
    atomic_add_f32(&cnt_out[r * N_NODES + src[i]], 1.0f);
    atomic_add_f32(&cnt_in [r * N_NODES + dst[i]], 1.0f);
}

// counts -> rsqrt(max(cnt,1)) in place; also total raw out-degree per node
__global__ __launch_bounds__(256)
void degtransform_kernel(float* __restrict__ rs_out, float* __restrict__ rs_in,
                         float* __restrict__ cnt_tot, int n) {
    int node = blockIdx.x * 256 + threadIdx.x;
    if (node >= n) return;
    float tot = 0.0f;
#pragma unroll
    for (int r = 0; r < N_REL; ++r) {
        float co = rs_out[r * N_NODES + node];
        tot += co;
        rs_out[r * N_NODES + node] = rsqrtf(fmaxf(co, 1.0f));
        float ci = rs_in[r * N_NODES + node];
        rs_in [r * N_NODES + node] = rsqrtf(fmaxf(ci, 1.0f));
    }
    cnt_tot[node] = tot;
}

// ---------------------------------------------------------------- bias init: hout[n][f] = sum_r b[r][f]
__global__ __launch_bounds__(256)
void biasinit_kernel(float* __restrict__ hout, const float* __restrict__ b, int total) {
    int i = blockIdx.x * 256 + threadIdx.x;
    if (i >= total) return;
    int f = i & (F - 1);
    hout[i] = b[f] + b[F + f] + b[2 * F + f] + b[3 * F + f];
}

// ---------------------------------------------------------------- edge gather + scatter-add (one wave per edge)
__global__ __launch_bounds__(256)
void edge_agg_kernel(float* __restrict__ agg, const float* __restrict__ x,
                     const int* __restrict__ src, const int* __restrict__ dst,
                     const float* __restrict__ rs_out, int n_edges) {
    int e = blockIdx.x * 8 + (threadIdx.x >> 5);
    if (e >= n_edges) return;
    int lane = threadIdx.x & 31;
    int s = src[e];
    int d = dst[e];
    float sc = rs_out[s];
    float4 v = ((const float4*)(x + (size_t)s * F))[lane];
    float* p = agg + (size_t)d * F + lane * 4;
    atomic_add_f32(p + 0, v.x * sc);
    atomic_add_f32(p + 1, v.y * sc);
    atomic_add_f32(p + 2, v.z * sc);
    atomic_add_f32(p + 3, v.w * sc);
}

// ---------------------------------------------------------------- fp32 WMMA GEMM: hout += (rs_in * agg) @ W
// block = 256 thr = 8 waves; each wave: one 16-row strip x full 128 cols (8 N-tiles).
// W staged in LDS pair-interleaved: Wl[(k>>1)*256 + 2n + (k&1)] so each B fragment
// ({W[kk][n], W[kk+1][n]}, kk even) is one aligned ds_load_b64 into an even reg pair.
__global__ __launch_bounds__(256)
void gemm_acc_kernel(float* __restrict__ hout, const float* __restrict__ agg,
                     const float* __restrict__ W, const float* __restrict__ rs_in,
                     int n_nodes) {
    __shared__ float Wl[F * F];  // 64 KB
    for (int i = threadIdx.x; i < (F * F) / 2; i += 256) {
        int kk2 = i >> 7;   // k-pair index 0..63
        int n   = i & 127;
        v2f w;
        w.x = W[(2 * kk2) * F + n];
        w.y = W[(2 * kk2 + 1) * F + n];
        ((v2f*)Wl)[kk2 * F + n] = w;
    }
    __syncthreads();

    const int wave = threadIdx.x >> 5;
    const int lane = threadIdx.x & 31;
    const int tile = (blockIdx.x * 8 + wave) * 16;  // wave-uniform
    if (tile >= n_nodes) return;
    const int l15 = lane & 15;
    const int hi  = lane >> 4;  // 0 or 1

    // A fragment row for this lane (clamped for the ragged tail)
    int arow = tile + l15;
    if (arow >= n_nodes) arow = n_nodes - 1;
    const float rsv = rs_in[arow];
    const float* arp = agg + (size_t)arow * F;

    // load current accumulation (bias / prior relations) per C/D layout
    v8f acc[8];
#pragma unroll
    for (int nt = 0; nt < 8; ++nt) {
#pragma unroll
        for (int v = 0; v < 8; ++v) {
            int row = tile + v + hi * 8;
            int rc = row < n_nodes ? row : n_nodes - 1;
            acc[nt][v] = hout[(size_t)rc * F + nt * 16 + l15];
        }
    }

    for (int k = 0; k < F; k += 4) {
        int kk = k + hi * 2;          // even
        v2f a;
        a.x = arp[kk]     * rsv;
        a.y = arp[kk + 1] * rsv;
        const v2f* brow = (const v2f*)Wl + (kk >> 1) * F;
#pragma unroll
        for (int nt = 0; nt < 8; ++nt) {
            v2f b = brow[nt * 16 + l15];
            acc[nt] = __builtin_amdgcn_wmma_f32_16x16x4_f32(
                false, a, false, b, (short)0, acc[nt], false, false);
        }
    }

#pragma unroll
    for (int nt = 0; nt < 8; ++nt) {
#pragma unroll
        for (int v = 0; v < 8; ++v) {
            int row = tile + v + hi * 8;
            if (row < n_nodes) hout[(size_t)row * F + nt * 16 + l15] = acc[nt][v];
        }
    }
}

// ---------------------------------------------------------------- fused l2-normalize + leaky relu (wave per row)
__global__ __launch_bounds__(256)
void norm_lrelu_kernel(float* __restrict__ h, int n) {
    int row = blockIdx.x * 8 + (threadIdx.x >> 5);
    if (row >= n) return;
    int lane = threadIdx.x & 31;
    float4* p = (float4*)(h + (size_t)row * F) + lane;
    float4 v = *p;
    float ss = v.x * v.x + v.y * v.y + v.z * v.z + v.w * v.w;
#pragma unroll
    for (int m = 16; m >= 1; m >>= 1) ss += __shfl_xor(ss, m, 32);
    float sc = 1.0f / fmaxf(sqrtf(ss), 1e-12f);
    float4 o;
    o.x = v.x * sc; o.y = v.y * sc; o.z = v.z * sc; o.w = v.w * sc;
    o.x = o.x > 0.0f ? o.x : 0.01f * o.x;
    o.y = o.y > 0.0f ? o.y : 0.01f * o.y;
    o.z = o.z > 0.0f ? o.z : 0.01f * o.z;
    o.w = o.w > 0.0f ? o.w : 0.01f * o.w;
    *p = o;
}

// ---------------------------------------------------------------- pooled mean: hg[f] = sum_n cnt_tot[n] * h[n][f]
__global__ __launch_bounds__(256)
void finalreduce_kernel(float* __restrict__ hg, const float* __restrict__ h,
                        const float* __restrict__ cnt_tot, int n) {
    int lane = threadIdx.x & 31;
    int wid = blockIdx.x * 8 + (threadIdx.x >> 5);
    int nw = gridDim.x * 8;
    float4 acc = {0.f, 0.f, 0.f, 0.f};
    for (int row = wid; row < n; row += nw) {
        float c = cnt_tot[row];
        float4 v = ((const float4*)(h + (size_t)row * F))[lane];
        acc.x += v.x * c; acc.y += v.y * c; acc.z += v.z * c; acc.w += v.w * c;
    }
    float* p = hg + lane * 4;
    atomic_add_f32(p + 0, acc.x);
    atomic_add_f32(p + 1, acc.y);
    atomic_add_f32(p + 2, acc.z);
    atomic_add_f32(p + 3, acc.w);
}

// ---------------------------------------------------------------- classifier: sigmoid(hg/N @ Wlin + blin)
__global__ void classifier_kernel(float* __restrict__ out, const float* __restrict__ hg,
                                  const float* __restrict__ Wlin, const float* __restrict__ blin) {
    int c = threadIdx.x;
    if (c >= N_CLASSES) return;
    float s = blin[c];
    const float invn = 1.0f / (float)N_NODES;
    for (int k = 0; k < F; ++k) s += hg[k] * invn * Wlin[k * N_CLASSES + c];
    out[c] = 1.0f / (1.0f + expf(-s));
}

// ================================================================ host
extern "C" void kernel_launch(void* const* d_in, const int* in_sizes, int n_in,
                              void* d_out, int out_size, void* d_ws, size_t ws_size,
                              hipStream_t stream) {
    const float* feat = (const float*)d_in[0];
    const int*   src  = (const int*)d_in[1];   // [4][800000]
    const int*   dst  = (const int*)d_in[2];
    const float* Wmat[3] = { (const float*)d_in[3], (const float*)d_in[5], (const float*)d_in[7] };
    const float* bvec[3] = { (const float*)d_in[4], (const float*)d_in[6], (const float*)d_in[8] };
    const float* Wlin = (const float*)d_in[9];
    const float* blin = (const float*)d_in[10];

    float* ws = (float*)d_ws;
    float* cnt_out = ws;                          // 4*N  (becomes rs_out)
    float* cnt_in  = cnt_out + 4 * N_NODES;       // 4*N  (becomes rs_in)
    float* cnt_tot = cnt_in  + 4 * N_NODES;       // N
    float* hg      = cnt_tot + N_NODES;           // 128
    float* bufA    = hg + F;                      // N*128
    float* bufB    = bufA + (size_t)N_NODES * F;  // N*128
    float* agg     = bufB + (size_t)N_NODES * F;  // N*128

    const int NF = N_NODES * F;

    // zero: counts + cnt_tot + hg (contiguous)
    {
        int zc = 4 * N_NODES * 2 + N_NODES + F;
        zero_kernel<<<(zc + 255) / 256, 256, 0, stream>>>(cnt_out, zc);
    }
    degree_kernel<<<(N_REL * N_EDGES + 255) / 256, 256, 0, stream>>>(
        cnt_out, cnt_in, src, dst, N_REL * N_EDGES);
    degtransform_kernel<<<(N_NODES + 255) / 256, 256, 0, stream>>>(
        cnt_out, cnt_in, cnt_tot, N_NODES);

    const float* x = feat;
    float* hout = bufA;
    for (int L = 0; L < 3; ++L) {
        biasinit_kernel<<<(NF + 255) / 256, 256, 0, stream>>>(hout, bvec[L], NF);
        for (int r = 0; r < N_REL; ++r) {
            zero_kernel<<<(NF + 255) / 256, 256, 0, stream>>>(agg, NF);
            edge_agg_kernel<<<(N_EDGES + 7) / 8, 256, 0, stream>>>(
                agg, x, src + r * N_EDGES, dst + r * N_EDGES,
                cnt_out + r * N_NODES, N_EDGES);
            gemm_acc_kernel<<<(N_NODES + 127) / 128, 256, 0, stream>>>(
                hout, agg, Wmat[L] + r * F * F, cnt_in + r * N_NODES, N_NODES);
        }
        if (L < 2)
            norm_lrelu_kernel<<<(N_NODES + 7) / 8, 256, 0, stream>>>(hout, N_NODES);
        x = hout;
        hout = (hout == bufA) ? bufB : bufA;
    }
    // x == bufA (layer-3 output)
    finalreduce_kernel<<<128, 256, 0, stream>>>(hg, x, cnt_tot, N_NODES);
    classifier_kernel<<<1, 32, 0, stream>>>((float*)d_out, hg, Wlin, blin);
}